// RuleLearner_80573586473469
// MI455X (gfx1250) — compile-verified
//
#include <hip/hip_runtime.h>
#include <hip/hip_bf16.h>

#define NI 4            // MAX_INVS
#define SL 5            // SEQ_LEN (= K)
#define NS 9            // 1 + NUM_SYMBOLS
#define CC (SL + NS)    // unary feature dim = 14
#define WPB 8           // waves (batch elements) per block
#define BLK (WPB * 32)

typedef __attribute__((ext_vector_type(2))) float v2f;
typedef __attribute__((ext_vector_type(8))) float v8f;

__device__ __forceinline__ float sigmf(float x) { return 1.0f / (1.0f + __expf(-x)); }

__global__ __launch_bounds__(BLK) void rule_learner_kernel(
    const int* __restrict__ inv_input, const int* __restrict__ seq_input,
    const int* __restrict__ inv_label, const float* __restrict__ inv_unaryp,
    const float* __restrict__ inv_binaryp, const float* __restrict__ inv_out_map,
    float* __restrict__ out, int batch)
{
  // inventor-side constants (shared by whole block)
  __shared__ float s_sp[20], s_ss[20];     // sigmoid(pos-cond), sigmoid(sym-cond) per (i,l)
  __shared__ float s_ib[20][SL];           // binary conds (masked) per (i,l,m)
  __shared__ float s_om[NI][SL + 1];       // softmaxed out_map
  __shared__ int   s_isym[20], s_label[NI];
  // per-wave scratch
  __shared__ float s_sat[WPB][20][SL];
  __shared__ float s_q[WPB][20];
  __shared__ float s_ns[WPB][20][SL];      // node_select (WMMA A-matrix source)
  __shared__ float s_coef[WPB][20];        // sel[i]*om[i][l] per row il

  const int tid  = threadIdx.x;
  const int lane = tid & 31;
  const int wv   = tid >> 5;

  // ---- per-block inventor-constant precompute (tiny, redundant per block) ----
  if (tid < 20) {
    const int i = tid / SL, l = tid % SL;
    const int sym = inv_input[tid];
    s_isym[tid] = sym;
    s_sp[tid] = sigmf(inv_unaryp[tid * CC + l]);            // position condition
    s_ss[tid] = sigmf(inv_unaryp[tid * CC + SL + sym]);     // symbol condition
    #pragma unroll
    for (int m = 0; m < SL; ++m) {
      const bool eq = (m != l) && (sym == inv_input[i * SL + m]);
      s_ib[tid][m] = eq ? sigmf(inv_binaryp[tid * SL + m]) : 0.0f;
    }
  } else if (tid < 20 + NI) {
    const int i = tid - 20;
    float v[SL + 1], mx = -1e30f;
    #pragma unroll
    for (int j = 0; j <= SL; ++j) { v[j] = inv_out_map[i * (SL + 1) + j]; mx = fmaxf(mx, v[j]); }
    float sum = 0.f;
    #pragma unroll
    for (int j = 0; j <= SL; ++j) { v[j] = __expf(v[j] - mx); sum += v[j]; }
    const float inv = 1.0f / sum;
    #pragma unroll
    for (int j = 0; j <= SL; ++j) s_om[i][j] = v[j] * inv;
    s_label[i] = inv_label[i];
  }
  __syncthreads();

  int b = blockIdx.x * WPB + wv;           // one wave per batch element
  const bool bvalid = (b < batch);
  if (!bvalid) b = 0;

  // broadcast the 5 sequence symbols across the wave
  int sv = (lane < SL) ? seq_input[b * SL + lane] : 0;
  const int sq0 = __shfl(sv, 0, 32), sq1 = __shfl(sv, 1, 32), sq2 = __shfl(sv, 2, 32),
            sq3 = __shfl(sv, 3, 32), sq4 = __shfl(sv, 4, 32);
  const int seqk[SL] = {sq0, sq1, sq2, sq3, sq4};

  const float SM = 8.0053682f;             // log(0.999*(I-1)/0.001)
  const int i_ = lane / SL, l_ = lane % SL;
  const bool act = (lane < 20);

  // ---- unary unification + binary sat (lane = (i, l) / (i, m)) ----
  float uni[SL];
  if (act) {
    const float sp = s_sp[lane], ss = s_ss[lane];
    const int mysym = s_isym[lane];
    #pragma unroll
    for (int k = 0; k < SL; ++k) {
      const float a = (k == l_) ? 1.0f : (1.0f - sp);
      const float c = (seqk[k] == mysym) ? 1.0f : (1.0f - ss);
      uni[k] = a * c;
    }
    #pragma unroll
    for (int k = 0; k < SL; ++k) {         // sat[i, m=l_, k] probsum over k2
      float p = 1.0f;
      #pragma unroll
      for (int k2 = 0; k2 < SL; ++k2)
        if (k2 != k) p *= (seqk[k2] == seqk[k]) ? (1.0f - uni[k2]) : 1.0f;
      s_sat[wv][lane][k] = 1.0f - p;
    }
  }
  __syncthreads();

  // ---- binary refinement, probsum-q, k-softmax (node_select) ----
  if (act) {
    float us[SL];
    #pragma unroll
    for (int k = 0; k < SL; ++k) {
      float f = uni[k];
      #pragma unroll
      for (int m = 0; m < SL; ++m)
        f *= 1.0f - s_ib[lane][m] * (1.0f - s_sat[wv][i_ * SL + m][k]);
      us[k] = f;
    }
    float pk = 1.0f;
    #pragma unroll
    for (int k = 0; k < SL; ++k) pk *= (1.0f - us[k]);
    s_q[wv][lane] = 1.0f - pk;
    float mx = -1e30f;
    #pragma unroll
    for (int k = 0; k < SL; ++k) mx = fmaxf(mx, us[k] * SM);
    float e[SL], sum = 0.f;
    #pragma unroll
    for (int k = 0; k < SL; ++k) { e[k] = __expf(us[k] * SM - mx); sum += e[k]; }
    const float inv = 1.0f / sum;
    #pragma unroll
    for (int k = 0; k < SL; ++k) s_ns[wv][lane][k] = e[k] * inv;
  }
  __syncthreads();

  // ---- inv_select (every lane, broadcast LDS reads) ----
  float iu[NI];
  #pragma unroll
  for (int i = 0; i < NI; ++i) {
    float p = 1.0f;
    #pragma unroll
    for (int l = 0; l < SL; ++l) p *= s_q[wv][i * SL + l];
    iu[i] = p;
  }
  float mxi = -1e30f;
  #pragma unroll
  for (int i = 0; i < NI; ++i) mxi = fmaxf(mxi, iu[i] * SM);
  float sel[NI], sums = 0.f;
  #pragma unroll
  for (int i = 0; i < NI; ++i) { sel[i] = __expf(iu[i] * SM - mxi); sums += sel[i]; }
  const float invs = 1.0f / sums;
  #pragma unroll
  for (int i = 0; i < NI; ++i) sel[i] *= invs;

  if (act) s_coef[wv][lane] = sel[i_] * s_om[i_][l_];   // row coefficient for D reduce

  // ---- WMMA: edge_outs[M=il (20, padded 32), N=u (9, padded 16)] =
  //            node_select[20x5] * onehot(seq)[5x16], K padded to 8 ----
  const int n = lane & 15;     // N column
  const int half = lane >> 4;  // lane half selects K pair (A) / K rows (B)
  const float m0 = (n < NS && sq0 == n) ? 1.0f : 0.0f;
  const float m1 = (n < NS && sq1 == n) ? 1.0f : 0.0f;
  const float m2 = (n < NS && sq2 == n) ? 1.0f : 0.0f;
  const float m3 = (n < NS && sq3 == n) ? 1.0f : 0.0f;
  const float m4 = (n < NS && sq4 == n) ? 1.0f : 0.0f;

  const int ka = half * 2;     // K base for this half (k-step 0)
  v2f a0, a1, bm;
  v8f d0 = {0,0,0,0,0,0,0,0}, d1 = {0,0,0,0,0,0,0,0};

  // k-step 0: K = 0..3
  a0.x = s_ns[wv][n][ka];  a0.y = s_ns[wv][n][ka + 1];                // rows 0..15
  a1.x = (n < 4) ? s_ns[wv][16 + n][ka]     : 0.0f;                   // rows 16..19
  a1.y = (n < 4) ? s_ns[wv][16 + n][ka + 1] : 0.0f;
  bm.x = half ? m2 : m0;
  bm.y = half ? m3 : m1;
  d0 = __builtin_amdgcn_wmma_f32_16x16x4_f32(false, a0, false, bm, (short)0, d0, false, false);
  d1 = __builtin_amdgcn_wmma_f32_16x16x4_f32(false, a1, false, bm, (short)0, d1, false, false);

  // k-step 1: K = 4 (K=5..7 zero padded)
  a0.x = half ? 0.0f : s_ns[wv][n][4];                   a0.y = 0.0f;
  a1.x = (!half && n < 4) ? s_ns[wv][16 + n][4] : 0.0f;  a1.y = 0.0f;
  bm.x = half ? 0.0f : m4;                               bm.y = 0.0f;
  d0 = __builtin_amdgcn_wmma_f32_16x16x4_f32(false, a0, false, bm, (short)0, d0, false, false);
  d1 = __builtin_amdgcn_wmma_f32_16x16x4_f32(false, a1, false, bm, (short)0, d1, false, false);

  __syncthreads();             // s_coef visible

  // ---- fused l- and i-reduction over D rows, butterfly across lane halves ----
  float acc = 0.0f;
  #pragma unroll
  for (int r = 0; r < 8; ++r)
    acc += s_coef[wv][r + 8 * half] * d0[r];             // rows 0..15
  if (half == 0) {
    #pragma unroll
    for (int r = 0; r < 4; ++r)
      acc += s_coef[wv][16 + r] * d1[r];                 // rows 16..19
  }
  acc += __shfl_xor(acc, 16, 32);                        // combine lane halves (same column)

  if (bvalid && half == 0 && n < NS) {
    float cu = 0.0f;                                     // out_map const-slot * label one-hot
    #pragma unroll
    for (int i = 0; i < NI; ++i)
      cu += sel[i] * s_om[i][SL] * ((s_label[i] == n) ? 1.0f : 0.0f);
    out[b * NS + n] = acc + cu;
  }
}

extern "C" void kernel_launch(void* const* d_in, const int* in_sizes, int n_in,
                              void* d_out, int out_size, void* d_ws, size_t ws_size,
                              hipStream_t stream) {
  (void)n_in; (void)out_size; (void)d_ws; (void)ws_size;
  const int*   inv_input   = (const int*)d_in[0];
  const int*   seq_input   = (const int*)d_in[1];
  const int*   inv_label   = (const int*)d_in[2];
  const float* inv_unaryp  = (const float*)d_in[3];
  const float* inv_binaryp = (const float*)d_in[4];
  const float* inv_out_map = (const float*)d_in[5];
  // d_in[6] = syms_eye: identity matrix, folded into the label-equality term.
  float* out = (float*)d_out;
  const int batch  = in_sizes[1] / SL;
  const int blocks = (batch + WPB - 1) / WPB;
  rule_learner_kernel<<<blocks, BLK, 0, stream>>>(
      inv_input, seq_input, inv_label, inv_unaryp, inv_binaryp, inv_out_map, out, batch);
}